// MyModel_43276090474803
// MI455X (gfx1250) — compile-verified
//
#include <hip/hip_runtime.h>
#include <math.h>

typedef float v2f __attribute__((ext_vector_type(2)));
typedef float v8f __attribute__((ext_vector_type(8)));

#define NROWS   131072
#define DDIM    64
#define ROWS_PER_BLOCK 128   // 8 waves * 16 rows

__global__ __launch_bounds__(256) void fused_pinet_kernel(
    const float* __restrict__ x,     // [B,64]
    const float* __restrict__ Wpi,   // [64,64]
    const float* __restrict__ Wsc,   // [64,1]
    const float* __restrict__ Ws0,   // [448,1]
    const float* __restrict__ Wp0,   // [448,1]
    const float* __restrict__ Ws1,   // [7,1]
    const float* __restrict__ Wp1,   // [7,1]
    const float* __restrict__ Wf,    // [6,1]
    float* __restrict__ out)         // [B,1]
{
    __shared__ float s_sum[8][16];
    __shared__ float s_prod[8][16];
    __shared__ float s_y1log[8][16];

    const int tid  = threadIdx.x;
    const int wave = tid >> 5;
    const int lane = tid & 31;
    const int half = lane >> 4;      // which 16-lane half
    const int m    = lane & 15;      // row-in-tile (A) / col-in-tile (B,C)
    const int rowBase = blockIdx.x * ROWS_PER_BLOCK + wave * 16;

    // ---- Load 16x64 tile of log|x| as WMMA A fragments; fold in shortcut dot
    const float* xrow = x + (size_t)(rowBase + m) * DDIM;
    v2f a[16];
    float ssc = 0.f;
#pragma unroll
    for (int kk = 0; kk < 16; ++kk) {
        const int k0 = 4 * kk + 2 * half;
        float2 v = *(const float2*)(xrow + k0);
        float l0 = __logf(fabsf(v.x));
        float l1 = __logf(fabsf(v.y));
        a[kk][0] = l0;
        a[kk][1] = l1;
        ssc = fmaf(l0, Wsc[k0],     ssc);
        ssc = fmaf(l1, Wsc[k0 + 1], ssc);
    }
    // complete row dot across the two halves -> log(y1) (shortcut product block)
    ssc += __shfl_xor(ssc, 16, 32);
    if (half == 0) s_y1log[wave][m] = ssc;

    // ---- Function-block partial sums per C-layout row slot
    float psum[8], pprod[8];
#pragma unroll
    for (int r = 0; r < 8; ++r) { psum[r] = 0.f; pprod[r] = 0.f; }

#pragma unroll
    for (int n = 0; n < 4; ++n) {
        const int col = 16 * n + m;

        // GEMM tile: z[r] = sum_k log|x|[row][k] * Wpi[k][col]
        v8f acc = {};
#pragma unroll
        for (int kk = 0; kk < 16; ++kk) {
            const int k0 = 4 * kk + 2 * half;
            v2f b;
            b[0] = Wpi[k0 * 64 + col];
            b[1] = Wpi[(k0 + 1) * 64 + col];
            acc = __builtin_amdgcn_wmma_f32_16x16x4_f32(
                false, a[kk], false, b, (short)0, acc, false, false);
        }

        // fb0 weights for this column (L0-resident)
        float ws[7], wp[7];
#pragma unroll
        for (int i = 0; i < 7; ++i) {
            ws[i] = Ws0[i * 64 + col];
            wp[i] = Wp0[i * 64 + col];
        }

#pragma unroll
        for (int r = 0; r < 8; ++r) {
            float z  = acc[r];                 // log(y0 element)
            float v  = __expf(z);              // y0 element, v > 0
            float t1 = __sinf(v);
            float t2 = __cosf(v);
            float t4 = __expf(v);
            float ei = __expf(-v);
            float e2 = t4 * t4;                // exp(2v); inf-safe below
            float t3 = 1.f - 2.f * __builtin_amdgcn_rcpf(e2 + 1.f);  // tanh(v)
            float t6 = __builtin_amdgcn_rcpf(1.f + ei);              // sigmoid(v)

            float s = psum[r];
            s = fmaf(v,  ws[0], s);
            s = fmaf(t1, ws[1], s);
            s = fmaf(t2, ws[2], s);
            s = fmaf(t3, ws[3], s);
            s = fmaf(t4, ws[4], s);
            s = fmaf(z,  ws[5], s);            // log|v| = z exactly
            s = fmaf(t6, ws[6], s);
            psum[r] = s;

            float p = pprod[r];
            p = fmaf(z,                  wp[0], p);   // log|v|       = z
            p = fmaf(__logf(fabsf(t1)),  wp[1], p);
            p = fmaf(__logf(fabsf(t2)),  wp[2], p);
            p = fmaf(__logf(fabsf(t3)),  wp[3], p);
            p = fmaf(v,                  wp[4], p);   // log|exp(v)|  = v
            p = fmaf(__logf(fabsf(z)),   wp[5], p);   // log|log|v||
            p = fmaf(__logf(t6),         wp[6], p);   // sigmoid > 0
            pprod[r] = p;
        }
    }

    // ---- Butterfly reduce across the 16 lanes of each half (per-row sums)
#pragma unroll
    for (int off = 1; off < 16; off <<= 1) {
#pragma unroll
        for (int r = 0; r < 8; ++r) {
            psum[r]  += __shfl_xor(psum[r],  off, 32);
            pprod[r] += __shfl_xor(pprod[r], off, 32);
        }
    }
    if (m == 0) {  // lane 0 -> rows 0..7, lane 16 -> rows 8..15
#pragma unroll
        for (int r = 0; r < 8; ++r) {
            s_sum[wave][8 * half + r]  = psum[r];
            s_prod[wave][8 * half + r] = pprod[r];
        }
    }
    __syncthreads();

    // ---- Per-row tail: fb1 on y1, concat/multiply, final dot (lanes 0..15)
    if (half == 0) {
        float y2s = s_sum[wave][m];
        float y2p = __expf(s_prod[wave][m]);
        float lu  = s_y1log[wave][m];      // log(y1)
        float u   = __expf(lu);            // y1 > 0

        float t1 = __sinf(u);
        float t2 = __cosf(u);
        float t4 = __expf(u);
        float ei = __expf(-u);
        float e2 = t4 * t4;
        float t3 = 1.f - 2.f * __builtin_amdgcn_rcpf(e2 + 1.f);  // tanh(u)
        float t6 = __builtin_amdgcn_rcpf(1.f + ei);              // sigmoid(u)

        float y3s = u * Ws1[0] + t1 * Ws1[1] + t2 * Ws1[2] + t3 * Ws1[3]
                  + t4 * Ws1[4] + lu * Ws1[5] + t6 * Ws1[6];

        float lp = 0.f;
        lp = fmaf(lu,                 Wp1[0], lp);  // log|u| = lu
        lp = fmaf(__logf(fabsf(t1)),  Wp1[1], lp);
        lp = fmaf(__logf(fabsf(t2)),  Wp1[2], lp);
        lp = fmaf(__logf(fabsf(t3)),  Wp1[3], lp);
        lp = fmaf(u,                  Wp1[4], lp);  // log|exp(u)| = u
        lp = fmaf(__logf(fabsf(lu)),  Wp1[5], lp);
        lp = fmaf(__logf(t6),         Wp1[6], lp);
        float y3p = __expf(lp);

        float r0 = y2s * Wf[0] + y2p * Wf[1] + y3s * Wf[2] + y3p * Wf[3]
                 + (y2s * y3s) * Wf[4] + (y2p * y3p) * Wf[5];

        out[rowBase + m] = r0;
    }
}

extern "C" void kernel_launch(void* const* d_in, const int* in_sizes, int n_in,
                              void* d_out, int out_size, void* d_ws, size_t ws_size,
                              hipStream_t stream) {
    const float* x   = (const float*)d_in[0];
    const float* Wpi = (const float*)d_in[1];
    const float* Wsc = (const float*)d_in[2];
    const float* Ws0 = (const float*)d_in[3];
    const float* Wp0 = (const float*)d_in[4];
    const float* Ws1 = (const float*)d_in[5];
    const float* Wp1 = (const float*)d_in[6];
    const float* Wf  = (const float*)d_in[7];
    float* out = (float*)d_out;

    const int blocks = NROWS / ROWS_PER_BLOCK;  // 1024
    fused_pinet_kernel<<<blocks, 256, 0, stream>>>(
        x, Wpi, Wsc, Ws0, Wp0, Ws1, Wp1, Wf, out);
}